// ROIBoxHead_39470749450916
// MI455X (gfx1250) — compile-verified
//
#include <hip/hip_runtime.h>
#include <cstdint>

// Problem constants (match reference)
#define C_COLS 1600
#define N_ROWS 8192
#define LVLS   4
#define KPL    256
#define MAXCH  5

// Rows staged in LDS per block (cls row = 6.4KB, box row = 25.6KB; both -> 51.2KB LDS)
#define CLS_R  8
#define BOX_R  2

// ---------------------------------------------------------------------------
// CDNA5 async global->LDS helpers
// ---------------------------------------------------------------------------
__device__ __forceinline__ void wait_async0() {
#if __has_builtin(__builtin_amdgcn_s_wait_asynccnt)
  __builtin_amdgcn_s_wait_asynccnt(0);
#else
  asm volatile("s_wait_asynccnt 0" ::: "memory");
#endif
}

// Copy n4 float4's from gbase (global, 16B aligned) to ldsbase (LDS, 16B aligned)
// using per-lane GLOBAL_LOAD_ASYNC_TO_LDS_B128 (GVS mode: saddr + 32-bit vgpr offset).
__device__ __forceinline__ void async_copy_f4(const float* gbase, float* ldsbase, int n4) {
  uint32_t lds0 = (uint32_t)(uintptr_t)ldsbase;       // low 32 bits of flat LDS ptr == LDS offset
  uint64_t g0   = (uint64_t)(uintptr_t)gbase;         // uniform -> SGPR pair
  for (int v = threadIdx.x; v < n4; v += 256) {
    uint32_t goff = (uint32_t)v << 4;
    uint32_t loff = lds0 + goff;
    asm volatile("global_load_async_to_lds_b128 %0, %1, %2"
                 :: "v"(loff), "v"(goff), "s"(g0)
                 : "memory");
  }
  wait_async0();        // each wave waits for its own async loads
  __syncthreads();      // then cross-wave visibility in LDS
}

// ---------------------------------------------------------------------------
// Chain builder: collapses the 4 sequential levels into per-column gather
// chains.  final[c] = sum of original columns chain[c][0..cnt-1].
//   cls: chain[tgt] = chain_prev[src] ++ [tgt]
//   box: chain[src] = chain_prev[tgt] ++ [src]
// Packed record: bits [11j+10 : 11j] = id_j (C=1600 < 2048), bits [57:55] = cnt.
// One block, 256 threads (== KPL). Levels separated by barriers so all reads
// of the previous level's state happen before any write of this level.
// ---------------------------------------------------------------------------
__global__ void build_chain_kernel(const int* __restrict__ tgt,
                                   const int* __restrict__ src,
                                   unsigned long long* __restrict__ packed,
                                   int isBox) {
  __shared__ int chain[C_COLS][MAXCH];   // 32.0 KB
  __shared__ int cnt[C_COLS];            //  6.4 KB
  for (int c = threadIdx.x; c < C_COLS; c += 256) {
    chain[c][0] = c;
    cnt[c] = 1;
  }
  __syncthreads();

  int tmp[MAXCH];
  for (int lvl = 0; lvl < LVLS; ++lvl) {
    int t = tgt[lvl * KPL + threadIdx.x];
    int s = src[lvl * KPL + threadIdx.x];
    int rd = isBox ? t : s;   // column read (previous level state)
    int wr = isBox ? s : t;   // column written (unique within a level)
    int n = cnt[rd];
    for (int j = 0; j < n; ++j) tmp[j] = chain[rd][j];
    __syncthreads();          // all reads done before any write
    for (int j = 0; j < n; ++j) chain[wr][j] = tmp[j];
    chain[wr][n] = wr;        // n <= 4 at level 3, so n+1 <= MAXCH
    cnt[wr] = n + 1;
    __syncthreads();
  }

  for (int c = threadIdx.x; c < C_COLS; c += 256) {
    int n = cnt[c];
    unsigned long long r = (unsigned long long)n << 55;
    for (int j = 0; j < n; ++j)
      r |= (unsigned long long)(chain[c][j] & 2047) << (11 * j);
    packed[c] = r;
  }
}

// ---------------------------------------------------------------------------
// cls branch: out[n][c] = sum_j logits[n][chain_c[j]]
// Block stages CLS_R full rows in LDS via async copy; thread owns columns
// {tid, tid+256, ...} with chain records preloaded into registers.
// ---------------------------------------------------------------------------
__global__ void __launch_bounds__(256)
cls_kernel(const float* __restrict__ logits,
           const unsigned long long* __restrict__ packed,
           float* __restrict__ out) {
  __shared__ __align__(16) float rows[CLS_R * C_COLS];   // 51.2 KB
  const int tid = threadIdx.x;
  const int n0  = blockIdx.x * CLS_R;

  unsigned long long rec[7];
#pragma unroll
  for (int s = 0; s < 7; ++s) {
    int c = s * 256 + tid;
    rec[s] = (c < C_COLS) ? packed[c] : 0ull;
  }

  async_copy_f4(logits + (size_t)n0 * C_COLS, rows, (CLS_R * C_COLS) / 4);

  for (int r = 0; r < CLS_R; ++r) {
    const float* row  = rows + r * C_COLS;
    float*       orow = out + (size_t)(n0 + r) * C_COLS;
#pragma unroll
    for (int s = 0; s < 7; ++s) {
      int c = s * 256 + tid;
      if (c < C_COLS) {
        unsigned long long t = rec[s];
        int n = (int)(t >> 55);
        float acc = row[(int)(t & 2047u)];
        for (int j = 1; j < n; ++j) {
          t >>= 11;
          acc += row[(int)(t & 2047u)];
        }
        orow[c] = acc;
      }
    }
  }
}

// ---------------------------------------------------------------------------
// box branch: out[n][c][q] = sum_j box[n][chainB_c[j]][q], q=0..3 -> float4 ops
// ---------------------------------------------------------------------------
struct __align__(16) f4 { float x, y, z, w; };

__global__ void __launch_bounds__(256)
box_kernel(const float* __restrict__ box,
           const unsigned long long* __restrict__ packed,
           float* __restrict__ out) {
  __shared__ __align__(16) float rows[BOX_R * 4 * C_COLS];  // 51.2 KB
  const int tid = threadIdx.x;
  const int n0  = blockIdx.x * BOX_R;

  unsigned long long rec[7];
#pragma unroll
  for (int s = 0; s < 7; ++s) {
    int c = s * 256 + tid;
    rec[s] = (c < C_COLS) ? packed[c] : 0ull;
  }

  async_copy_f4(box + (size_t)n0 * 4 * C_COLS, rows, (BOX_R * 4 * C_COLS) / 4);

  for (int r = 0; r < BOX_R; ++r) {
    const f4* row4 = (const f4*)(rows + r * 4 * C_COLS);
    f4*       orow = (f4*)(out + (size_t)(n0 + r) * 4 * C_COLS);
#pragma unroll
    for (int s = 0; s < 7; ++s) {
      int c = s * 256 + tid;
      if (c < C_COLS) {
        unsigned long long t = rec[s];
        int n = (int)(t >> 55);
        f4 acc = row4[(int)(t & 2047u)];
        for (int j = 1; j < n; ++j) {
          t >>= 11;
          f4 v = row4[(int)(t & 2047u)];
          acc.x += v.x; acc.y += v.y; acc.z += v.z; acc.w += v.w;
        }
        orow[c] = acc;   // coalesced b128 store
      }
    }
  }
}

// ---------------------------------------------------------------------------
extern "C" void kernel_launch(void* const* d_in, const int* in_sizes, int n_in,
                              void* d_out, int out_size, void* d_ws, size_t ws_size,
                              hipStream_t stream) {
  const float* class_logits   = (const float*)d_in[0];
  const float* box_regression = (const float*)d_in[1];
  const int*   label_targets  = (const int*)d_in[2];
  const int*   label_sources  = (const int*)d_in[3];

  float* out_cls = (float*)d_out;
  float* out_box = out_cls + (size_t)N_ROWS * C_COLS;

  unsigned long long* clsPacked = (unsigned long long*)d_ws;       // 12.8 KB
  unsigned long long* boxPacked = clsPacked + C_COLS;              // 12.8 KB

  build_chain_kernel<<<1, 256, 0, stream>>>(label_targets, label_sources, clsPacked, 0);
  build_chain_kernel<<<1, 256, 0, stream>>>(label_targets, label_sources, boxPacked, 1);

  cls_kernel<<<N_ROWS / CLS_R, 256, 0, stream>>>(class_logits, clsPacked, out_cls);
  box_kernel<<<N_ROWS / BOX_R, 256, 0, stream>>>(box_regression, boxPacked, out_box);
}